// ReaAllocation_47931835023416
// MI455X (gfx1250) — compile-verified
//
#include <hip/hip_runtime.h>
#include <math.h>

typedef float v2f __attribute__((ext_vector_type(2)));
typedef float v8f __attribute__((ext_vector_type(8)));

#define RTR 8
#define IN_DIM 384
#define HID 64
#define BB 8192
#define EE 1024
#define AUX_COEF 0.05f
#define EPSN 1e-12f

__device__ __forceinline__ v8f wmma_f32_4(v2f a, v2f b, v8f c) {
  // D = A(16x4,f32) * B(4x16,f32) + C(16x16,f32)
  return __builtin_amdgcn_wmma_f32_16x16x4_f32(false, a, false, b, (short)0, c,
                                               false, false);
}

// ---------------------------------------------------------------------------
// Gate: logits = x@gate_w + gate_b ; softmax probs ; top-2 ; gate weights ;
// atomic sums for aux loss. One thread per sample.
// ---------------------------------------------------------------------------
__global__ void gate_kernel(const float* __restrict__ x,
                            const float* __restrict__ gw,
                            const float* __restrict__ gb,
                            float* __restrict__ stats,     // [16]: probs_sum, mask_sum
                            float* __restrict__ wpr) {     // [B][R] gate weight per router
  int b = blockIdx.x * blockDim.x + threadIdx.x;
  if (b >= BB) return;
  float acc[RTR];
#pragma unroll
  for (int r = 0; r < RTR; ++r) acc[r] = gb[r];
  const float* xr = x + (size_t)b * IN_DIM;
  for (int d = 0; d < IN_DIM; ++d) {
    float xv = xr[d];
    const float* g = gw + (size_t)d * RTR;
#pragma unroll
    for (int r = 0; r < RTR; ++r) acc[r] = fmaf(xv, g[r], acc[r]);
  }
  // softmax over 8
  float mx = acc[0];
#pragma unroll
  for (int r = 1; r < RTR; ++r) mx = fmaxf(mx, acc[r]);
  float p[RTR], s = 0.f;
#pragma unroll
  for (int r = 0; r < RTR; ++r) { p[r] = expf(acc[r] - mx); s += p[r]; }
  float inv = 1.0f / s;
  // top-2 (earliest index wins ties, as in lax.top_k)
  int i1 = 0;
#pragma unroll
  for (int r = 1; r < RTR; ++r) if (acc[r] > acc[i1]) i1 = r;
  int i2 = (i1 == 0) ? 1 : 0;
#pragma unroll
  for (int r = 0; r < RTR; ++r) if (r != i1 && acc[r] > acc[i2]) i2 = r;
  float e2 = expf(acc[i2] - acc[i1]);
  float den = 1.0f + e2;
  float g1 = 1.0f / den, g2 = e2 / den;
#pragma unroll
  for (int r = 0; r < RTR; ++r) {
    wpr[(size_t)b * RTR + r] = (r == i1) ? g1 : ((r == i2) ? g2 : 0.0f);
    atomicAdd(&stats[r], p[r] * inv);
  }
  atomicAdd(&stats[RTR + i1], 1.0f);
  atomicAdd(&stats[RTR + i2], 1.0f);
}

__global__ void aux_kernel(const float* __restrict__ stats, float* __restrict__ out_aux) {
  float a = 0.f;
#pragma unroll
  for (int r = 0; r < RTR; ++r)
    a += (stats[r] / (float)BB) * (stats[RTR + r] / (float)BB);
  *out_aux = (float)RTR * a * AUX_COEF;
}

// ---------------------------------------------------------------------------
// out[r][row][h] = l2norm_row( src[row,:] @ W[r][:, :] + bias[r][:] )
// One wave per (router, 16-row tile). fp32 WMMA 16x16x4 over K=384, 4 H-tiles.
// A layout: lane L: M = L&15, VGPRs hold K = k0 + 2*(L>>4) + {0,1}
// B layout: lane L: N = L&15, VGPRs hold K = k0 + 2*(L>>4) + {0,1}
// C layout: VGPR j: M = j + 8*(L>>4), N = L&15
// ---------------------------------------------------------------------------
__global__ void proj_norm_kernel(const float* __restrict__ src,
                                 const float* __restrict__ W,
                                 const float* __restrict__ bias,
                                 float* __restrict__ out, int N) {
  int r = blockIdx.y;
  int row0 = blockIdx.x * 16;
  int lane = threadIdx.x & 31;
  int lx = lane & 15;
  int half = lane >> 4;
  const float* Wr = W + (size_t)r * IN_DIM * HID;
  const float* arow = src + (size_t)(row0 + lx) * IN_DIM;

  v8f c[4] = {};
  for (int k0 = 0; k0 < IN_DIM; k0 += 4) {
    int ka = k0 + 2 * half;
    v2f a;
    a.x = arow[ka];
    a.y = arow[ka + 1];
#pragma unroll
    for (int t = 0; t < 4; ++t) {
      int h0 = t * 16;
      v2f bf;
      bf.x = Wr[(size_t)ka * HID + h0 + lx];
      bf.y = Wr[(size_t)(ka + 1) * HID + h0 + lx];
      c[t] = wmma_f32_4(a, bf, c[t]);
    }
  }

  __shared__ float tile[16][HID];
  __shared__ float rno[16];
#pragma unroll
  for (int t = 0; t < 4; ++t)
#pragma unroll
    for (int j = 0; j < 8; ++j)
      tile[j + 8 * half][t * 16 + lx] = c[t][j] + bias[r * HID + t * 16 + lx];
  __syncthreads();
  if (lane < 16) {
    float sq = 0.f;
#pragma unroll
    for (int h = 0; h < HID; ++h) { float v = tile[lane][h]; sq += v * v; }
    rno[lane] = 1.0f / fmaxf(sqrtf(sq), EPSN);
  }
  __syncthreads();
  float* orow = out + ((size_t)r * N + row0) * HID;
  for (int i = lane; i < 16 * HID; i += 32) {
    int m = i >> 6, h = i & 63;
    orow[(size_t)m * HID + h] = tile[m][h] * rno[m];
  }
}

// ---------------------------------------------------------------------------
// Per 16-sample tile: for each router r with nonzero gate weight in the tile,
// scores[16,1024] = xu[r]·ev[r]^T via fp32 WMMA (kept fully in registers),
// softmax over E (online max/sum, shuffle + LDS combine), gate-weighted
// accumulate into 64 regs/lane. Then write rea_probs rows and do the
// inverse-CDF sample + log-prob.
// 256 threads = 8 waves; wave w owns expert tiles e0 = 16*(w + 8*t), t=0..7.
// The 16x64 xu tile is staged into LDS once per router with
// GLOBAL_LOAD_ASYNC_TO_LDS_B128 (ASYNCcnt), removing the 8x-redundant
// per-wave global reads of the WMMA A operand.
// ---------------------------------------------------------------------------
__global__ void moe_kernel(const float* __restrict__ xu,
                           const float* __restrict__ ev,
                           const float* __restrict__ wpr,
                           const float* __restrict__ rnd,
                           float* __restrict__ rea,
                           float* __restrict__ out_idx,
                           float* __restrict__ out_lp) {
  int b0 = blockIdx.x * 16;
  int tid = threadIdx.x;
  int wv = tid >> 5;
  int lane = tid & 31;
  int lx = lane & 15;
  int half = lane >> 4;
  int mrow = 8 * half;  // C vgpr j -> row m = j + mrow

  __shared__ float xuTile[16 * HID];  // 4KB A tile, async-staged per router
  __shared__ float partM[8][16];
  __shared__ float partS[8][16];
  __shared__ float rowM[16], rowS[16];
  __shared__ float wrow_s[16];
  __shared__ int anyw;

  float accp[8][8];  // [t][j] weighted prob accumulator
#pragma unroll
  for (int t = 0; t < 8; ++t)
#pragma unroll
    for (int j = 0; j < 8; ++j) accp[t][j] = 0.f;

  for (int r = 0; r < RTR; ++r) {
    __syncthreads();  // previous iteration's LDS readers done
    if (tid == 0) {
      int a = 0;
      for (int m = 0; m < 16; ++m) {
        float w = wpr[(size_t)(b0 + m) * RTR + r];
        wrow_s[m] = w;
        if (w != 0.f) a = 1;
      }
      anyw = a;
    }
    // --- async-stage the 16x64 xu[r] tile into LDS: 16B per thread
    {
      int elem = tid * 4;  // float index, 256 threads * 4 floats = 1024
      const float* gsrc = xu + ((size_t)r * BB + b0) * HID + elem;
      unsigned lds_off = (unsigned)(uintptr_t)(&xuTile[elem]);
      asm volatile("global_load_async_to_lds_b128 %0, %1, off"
                   :
                   : "v"(lds_off), "v"(gsrc)
                   : "memory");
      asm volatile("s_wait_asynccnt 0x0" ::: "memory");
    }
    __syncthreads();
    if (!anyw) continue;  // block-uniform: EXEC stays all-ones for WMMA

    // --- scores for this router: 8 expert tiles per wave, K=64 in steps of 4
    const float* evr = ev + (size_t)r * EE * HID;
    v8f c[8] = {};
    for (int k0 = 0; k0 < HID; k0 += 4) {
      int ka = k0 + 2 * half;
      v2f a;
      a.x = xuTile[lx * HID + ka];      // ds_load from staged tile
      a.y = xuTile[lx * HID + ka + 1];
#pragma unroll
      for (int t = 0; t < 8; ++t) {
        int e0 = 16 * (wv + 8 * t);
        const float* erow = evr + (size_t)(e0 + lx) * HID + ka;
        v2f bf;
        bf.x = erow[0];
        bf.y = erow[1];
        c[t] = wmma_f32_4(a, bf, c[t]);
      }
    }

    // --- online softmax stats: per-lane over its 8 tiles
    float ml[8], sl[8];
#pragma unroll
    for (int j = 0; j < 8; ++j) {
      float m = c[0][j];
#pragma unroll
      for (int t = 1; t < 8; ++t) m = fmaxf(m, c[t][j]);
      float s = 0.f;
#pragma unroll
      for (int t = 0; t < 8; ++t) s += expf(c[t][j] - m);
      ml[j] = m;
      sl[j] = s;
    }
    // combine across the 16 lanes that share each row (width-16 xor shuffles)
#pragma unroll
    for (int off = 8; off >= 1; off >>= 1) {
#pragma unroll
      for (int j = 0; j < 8; ++j) {
        float om = __shfl_xor(ml[j], off, 16);
        float os = __shfl_xor(sl[j], off, 16);
        float nm = fmaxf(ml[j], om);
        sl[j] = sl[j] * expf(ml[j] - nm) + os * expf(om - nm);
        ml[j] = nm;
      }
    }
    if (lx == 0) {
#pragma unroll
      for (int j = 0; j < 8; ++j) {
        partM[wv][mrow + j] = ml[j];
        partS[wv][mrow + j] = sl[j];
      }
    }
    __syncthreads();
    if (tid < 16) {
      float m = partM[0][tid], s = partS[0][tid];
#pragma unroll
      for (int w2 = 1; w2 < 8; ++w2) {
        float om = partM[w2][tid], os = partS[w2][tid];
        float nm = fmaxf(m, om);
        s = s * expf(m - nm) + os * expf(om - nm);
        m = nm;
      }
      rowM[tid] = m;
      rowS[tid] = s;
    }
    __syncthreads();

    // --- gate-weighted prob accumulate (mapping identical every router)
#pragma unroll
    for (int j = 0; j < 8; ++j) {
      int m = mrow + j;
      float winv = wrow_s[m] / rowS[m];
      float rm = rowM[m];
#pragma unroll
      for (int t = 0; t < 8; ++t)
        accp[t][j] += winv * expf(c[t][j] - rm);
    }
  }

  // --- write this tile's rea_probs rows
#pragma unroll
  for (int t = 0; t < 8; ++t) {
    int e0 = 16 * (wv + 8 * t);
#pragma unroll
    for (int j = 0; j < 8; ++j) {
      int m = mrow + j;
      rea[(size_t)(b0 + m) * EE + e0 + lx] = accp[t][j];
    }
  }
  __threadfence();
  __syncthreads();

  // --- inverse-CDF categorical sample + log-prob (1 thread per row)
  if (tid < 16) {
    int b = b0 + tid;
    float rv = rnd[b];
    const float* row = rea + (size_t)b * EE;
    float cs = 0.f;
    int idx = -1;
    float pv = 0.f;
    for (int e = 0; e < EE; ++e) {
      float v = row[e];
      cs += v;
      if (idx < 0 && cs > rv) { idx = e; pv = v; }
    }
    if (idx < 0) { idx = 0; pv = row[0]; }  // argmax of all-False -> 0
    out_idx[b] = (float)idx;
    out_lp[b] = logf(pv);
  }
}

// ---------------------------------------------------------------------------
extern "C" void kernel_launch(void* const* d_in, const int* in_sizes, int n_in,
                              void* d_out, int out_size, void* d_ws, size_t ws_size,
                              hipStream_t stream) {
  const float* x   = (const float*)d_in[0];  // [B, 384]
  const float* emb = (const float*)d_in[1];  // [E, 384]
  const float* rnd = (const float*)d_in[2];  // [B, 1]
  const float* gw  = (const float*)d_in[3];  // [384, 8]
  const float* gb  = (const float*)d_in[4];  // [8]
  const float* Uw  = (const float*)d_in[5];  // [8, 384, 64]
  const float* Ub  = (const float*)d_in[6];  // [8, 64]
  const float* Vw  = (const float*)d_in[7];  // [8, 384, 64]
  const float* Vb  = (const float*)d_in[8];  // [8, 64]
  (void)in_sizes; (void)n_in; (void)out_size; (void)ws_size;

  float* ws    = (float*)d_ws;
  float* stats = ws;                                   // 16
  float* wpr   = stats + 16;                           // B*R      = 65536
  float* evb   = wpr + (size_t)BB * RTR;               // R*E*H    = 524288
  float* xub   = evb + (size_t)RTR * EE * HID;         // R*B*H    = 4194304
  float* rea   = xub + (size_t)RTR * BB * HID;         // B*E      = 8388608
  float* out   = (float*)d_out;  // [0,B): selected_index, [B,2B): log_probs, [2B]: aux

  hipMemsetAsync(stats, 0, 16 * sizeof(float), stream);
  gate_kernel<<<BB / 256, 256, 0, stream>>>(x, gw, gb, stats, wpr);
  aux_kernel<<<1, 1, 0, stream>>>(stats, out + 2 * BB);
  proj_norm_kernel<<<dim3(EE / 16, RTR), 32, 0, stream>>>(emb, Vw, Vb, evb, EE);
  proj_norm_kernel<<<dim3(BB / 16, RTR), 32, 0, stream>>>(x, Uw, Ub, xub, BB);
  moe_kernel<<<BB / 16, 256, 0, stream>>>(xub, evb, wpr, rnd, rea, out, out + BB);
}